// RobustInfoNCELoss_35150012350941
// MI455X (gfx1250) — compile-verified
//
#include <hip/hip_runtime.h>

#define B_N 8192
#define D_K 128
#define INV_T 14.285714285714286f
#define NTILES (B_N / 16)      // 512 column tiles
#define LDS_STRIDE 132         // padded row stride (floats), multiple of 4, bank-spread

typedef __attribute__((ext_vector_type(2))) float v2f;
typedef __attribute__((ext_vector_type(8))) float v8f;

// ---------------------------------------------------------------------------
// Kernel 1: L2-normalize rows of z_ts and z_tab (one wave per row), zero accum
// ---------------------------------------------------------------------------
__global__ __launch_bounds__(256) void norm_kernel(const float* __restrict__ zts,
                                                   const float* __restrict__ ztab,
                                                   float* __restrict__ ots,
                                                   float* __restrict__ otab,
                                                   float* __restrict__ accum) {
  int gtid = blockIdx.x * 256 + threadIdx.x;
  int wave = gtid >> 5;
  int lane = threadIdx.x & 31;
  const float* src;
  float* dst;
  int row;
  if (wave < B_N) { src = zts;  dst = ots;  row = wave; }
  else            { src = ztab; dst = otab; row = wave - B_N; }

  float4 v = ((const float4*)(src + (size_t)row * D_K))[lane];
  float ss = v.x * v.x + v.y * v.y + v.z * v.z + v.w * v.w;
#pragma unroll
  for (int o = 16; o >= 1; o >>= 1) ss += __shfl_xor(ss, o, 32);
  float inv = 1.0f / fmaxf(sqrtf(ss), 1e-12f);
  float4 r;
  r.x = v.x * inv; r.y = v.y * inv; r.z = v.z * inv; r.w = v.w * inv;
  ((float4*)(dst + (size_t)row * D_K))[lane] = r;

  if (gtid == 0) { accum[0] = 0.0f; accum[1] = 0.0f; }
}

// ---------------------------------------------------------------------------
// 16x16 tile GEMM over K=128 using fp32 WMMA (16x16x4), two row sub-tiles
// sharing the B fragment.
// ---------------------------------------------------------------------------
__device__ __forceinline__ void wmma_tile(const float* __restrict__ ap0,
                                          const float* __restrict__ ap1,
                                          const float* __restrict__ bp,
                                          v8f& c0, v8f& c1) {
  v8f acc0 = {};
  v8f acc1 = {};
#pragma unroll
  for (int k = 0; k < D_K; k += 4) {
    v2f a0 = *(const v2f*)(ap0 + k);
    v2f a1 = *(const v2f*)(ap1 + k);
    v2f b  = *(const v2f*)(bp + k);
    acc0 = __builtin_amdgcn_wmma_f32_16x16x4_f32(false, a0, false, b, (short)0,
                                                 acc0, false, false);
    acc1 = __builtin_amdgcn_wmma_f32_16x16x4_f32(false, a1, false, b, (short)0,
                                                 acc1, false, false);
  }
  c0 = acc0;
  c1 = acc1;
}

// ---------------------------------------------------------------------------
// Kernel 2: fused flash-style masked-softmax loss.
// One block = 32 rows (two 16-row WMMA sub-tiles). 8 waves stride over the
// 512 column tiles. Pass 1: row max. Pass 2: exp-sum / masked sum / counts.
// ---------------------------------------------------------------------------
__global__ __launch_bounds__(256) void loss_kernel(const float* __restrict__ Am,
                                                   const float* __restrict__ Bm,
                                                   const unsigned char* __restrict__ mask,
                                                   float* __restrict__ accum) {
  __shared__ float lds_a[32 * LDS_STRIDE];
  __shared__ float smax[8][32];
  __shared__ float rowM[32];
  __shared__ float sred[8][32][3];

  const int tid  = threadIdx.x;
  const int wave = tid >> 5;
  const int lane = tid & 31;
  const int hlf  = lane >> 4;   // half-wave: selects K-pair
  const int l16  = lane & 15;   // row (A) / col (B) within tile
  const int rowBase = blockIdx.x * 32;

  // Stage 32x128 A tile into LDS (padded stride).
  for (int idx = tid; idx < 32 * (D_K / 4); idx += 256) {
    int r = idx >> 5;          // 32 float4 per row
    int q = idx & 31;
    float4 v = ((const float4*)(Am + (size_t)(rowBase + r) * D_K))[q];
    float* d = &lds_a[r * LDS_STRIDE + q * 4];
    d[0] = v.x; d[1] = v.y; d[2] = v.z; d[3] = v.w;
  }
  __syncthreads();

  const float* ap0 = &lds_a[l16 * LDS_STRIDE + hlf * 2];
  const float* ap1 = &lds_a[(16 + l16) * LDS_STRIDE + hlf * 2];

  // ---------------- Pass 1: per-row max (in dot space) ----------------
  float m0[8], m1[8];
#pragma unroll
  for (int v = 0; v < 8; ++v) { m0[v] = -INFINITY; m1[v] = -INFINITY; }

  for (int t = wave; t < NTILES; t += 8) {
    const float* bp = Bm + (size_t)(t * 16 + l16) * D_K + hlf * 2;
    v8f c0, c1;
    wmma_tile(ap0, ap1, bp, c0, c1);
#pragma unroll
    for (int v = 0; v < 8; ++v) {
      m0[v] = fmaxf(m0[v], c0[v]);
      m1[v] = fmaxf(m1[v], c1[v]);
    }
  }
#pragma unroll
  for (int v = 0; v < 8; ++v) {
#pragma unroll
    for (int o = 8; o >= 1; o >>= 1) {
      m0[v] = fmaxf(m0[v], __shfl_xor(m0[v], o, 32));
      m1[v] = fmaxf(m1[v], __shfl_xor(m1[v], o, 32));
    }
  }
  if (l16 == 0) {
#pragma unroll
    for (int v = 0; v < 8; ++v) {
      smax[wave][hlf * 8 + v]      = m0[v];
      smax[wave][16 + hlf * 8 + v] = m1[v];
    }
  }
  __syncthreads();
  if (tid < 32) {
    float mm = smax[0][tid];
#pragma unroll
    for (int w = 1; w < 8; ++w) mm = fmaxf(mm, smax[w][tid]);
    rowM[tid] = mm;
  }
  __syncthreads();

  float M0[8], M1[8];
#pragma unroll
  for (int v = 0; v < 8; ++v) {
    M0[v] = rowM[hlf * 8 + v] * INV_T;        // row max in logit space
    M1[v] = rowM[16 + hlf * 8 + v] * INV_T;
  }

  // ---------------- Pass 2: exp-sum, masked sum, counts ----------------
  float e0[8], e1[8], p0[8], p1[8], n0[8], n1[8];
#pragma unroll
  for (int v = 0; v < 8; ++v) {
    e0[v] = e1[v] = p0[v] = p1[v] = n0[v] = n1[v] = 0.0f;
  }

  for (int t = wave; t < NTILES; t += 8) {
    const int colBase = t * 16;
    const float* bp = Bm + (size_t)(colBase + l16) * D_K + hlf * 2;
    v8f c0, c1;
    wmma_tile(ap0, ap1, bp, c0, c1);
    const int col = colBase + l16;
#pragma unroll
    for (int v = 0; v < 8; ++v) {
      float s0 = c0[v] * INV_T;
      float s1 = c1[v] * INV_T;
      e0[v] += __expf(s0 - M0[v]);
      e1[v] += __expf(s1 - M1[v]);
      int r0 = rowBase + hlf * 8 + v;
      int r1 = r0 + 16;
      unsigned char k0 = mask[(size_t)r0 * B_N + col];
      unsigned char k1 = mask[(size_t)r1 * B_N + col];
      if (k0) { p0[v] += s0; n0[v] += 1.0f; }
      if (k1) { p1[v] += s1; n1[v] += 1.0f; }
    }
  }

#pragma unroll
  for (int v = 0; v < 8; ++v) {
#pragma unroll
    for (int o = 8; o >= 1; o >>= 1) {
      e0[v] += __shfl_xor(e0[v], o, 32);
      e1[v] += __shfl_xor(e1[v], o, 32);
      p0[v] += __shfl_xor(p0[v], o, 32);
      p1[v] += __shfl_xor(p1[v], o, 32);
      n0[v] += __shfl_xor(n0[v], o, 32);
      n1[v] += __shfl_xor(n1[v], o, 32);
    }
  }
  if (l16 == 0) {
#pragma unroll
    for (int v = 0; v < 8; ++v) {
      int r0 = hlf * 8 + v, r1 = 16 + hlf * 8 + v;
      sred[wave][r0][0] = e0[v]; sred[wave][r0][1] = p0[v]; sred[wave][r0][2] = n0[v];
      sred[wave][r1][0] = e1[v]; sred[wave][r1][1] = p1[v]; sred[wave][r1][2] = n1[v];
    }
  }
  __syncthreads();

  if (tid < 32) {
    float L = 0.0f, PS = 0.0f, CNT = 0.0f;
#pragma unroll
    for (int w = 0; w < 8; ++w) {
      L   += sred[w][tid][0];
      PS  += sred[w][tid][1];
      CNT += sred[w][tid][2];
    }
    float M = rowM[tid] * INV_T;
    float logden = logf(L + 1e-8f);
    bool valid = CNT > 0.0f;
    // sum mask*(s - M) = PS - M*CNT
    float per = -((PS - M * CNT) / fmaxf(CNT, 1.0f) - logden);
    float lossC = valid ? per : 0.0f;
    float vldC  = valid ? 1.0f : 0.0f;
#pragma unroll
    for (int o = 16; o >= 1; o >>= 1) {
      lossC += __shfl_xor(lossC, o, 32);
      vldC  += __shfl_xor(vldC, o, 32);
    }
    if (tid == 0) {
      atomicAdd(&accum[0], lossC);
      atomicAdd(&accum[1], vldC);
    }
  }
}

// ---------------------------------------------------------------------------
// Kernel 3: finalize scalar
// ---------------------------------------------------------------------------
__global__ void final_kernel(const float* __restrict__ accum, float* __restrict__ out) {
  out[0] = (accum[1] > 0.0f) ? (accum[0] / accum[1]) : 0.0f;
}

// ---------------------------------------------------------------------------
extern "C" void kernel_launch(void* const* d_in, const int* in_sizes, int n_in,
                              void* d_out, int out_size, void* d_ws, size_t ws_size,
                              hipStream_t stream) {
  const float* zts  = (const float*)d_in[0];
  const float* ztab = (const float*)d_in[1];
  const unsigned char* mask = (const unsigned char*)d_in[2];  // jax bool_ = 1 byte

  float* ws    = (float*)d_ws;
  float* ztsn  = ws;                                  // 8192*128 f32
  float* ztabn = ws + (size_t)B_N * D_K;              // 8192*128 f32
  float* accum = ws + 2 * (size_t)B_N * D_K;          // 2 f32
  float* out   = (float*)d_out;

  norm_kernel<<<(2 * B_N) / 8, 256, 0, stream>>>(zts, ztab, ztsn, ztabn, accum);
  loss_kernel<<<B_N / 32, 256, 0, stream>>>(ztsn, ztabn, mask, accum);
  final_kernel<<<1, 1, 0, stream>>>(accum, out);
}